// FieldLayer_61383672594915
// MI455X (gfx1250) — compile-verified
//
#include <hip/hip_runtime.h>
#include <hip/hip_bf16.h>
#include <math.h>

// FieldLayer for MI455X (gfx1250, wave32).
//
// Strategy: the reference is linear up to the gates, so the three weight
// stages fold into two fused matrices computed once per launch:
//   W0 = W_tp_0e @ W_lin_0e @ W_sc_0e   (32x64), scale C0 = 1/(64*sqrt(96))
//   W1 = W_tp_1o @ W_lin_1o @ W_sc_1o   (64x32), scale C1 = 1/256
// Then per node:
//   s1[u]      = sum_m x1[u][m]*field[m]
//   z0         = C0 * a * (s1 @ W0)           -> r0 = x0 + z0
//   z1[w][m]   = C1 * a * field[m]*(x0@W1)[w] -> r1 = x1 + z1
//   gate: o = r * (ssp(|r|)/|r|), ssp(x) = softplus(x) - log(2)
//
// ~1.3 KB moved per node vs ~8.3 KFLOP -> HBM-bound (~11us at 23.3 TB/s).
// GEMMs run on V_WMMA_F32_16X16X4_F32 (full fp32 matrix op): one wave per
// 16-node tile, fused weights staged transposed in LDS so each B fragment
// is a single 8-byte ds_load.

#define MUL0 64
#define MUL1 32
#define ROWLEN 160              // MUL0 + 3*MUL1
#define LOG2C 0.6931471805599453f
#define EPS2 (1e-8f * 1e-8f)
#define TILES_PER_BLOCK 8       // 8 waves * 32 lanes = 256 threads

typedef __attribute__((ext_vector_type(2))) float v2f;
typedef __attribute__((ext_vector_type(8))) float v8f;

// ---------------------------------------------------------------------------
// Kernel 1: fold the three weight stages into W0T (64x32, [w][u]) and
// W1T (32x64, [w][u]) in workspace, scales included. Single 256-thread block.
// ---------------------------------------------------------------------------
__global__ __launch_bounds__(256) void fuse_weights_kernel(
    const float* __restrict__ Wtp0,  // (32,64)
    const float* __restrict__ Wtp1,  // (64,32)
    const float* __restrict__ Wlin0, // (64,64)
    const float* __restrict__ Wlin1, // (32,32)
    const float* __restrict__ Wsc0,  // (64,64)
    const float* __restrict__ Wsc1,  // (32,32)
    float* __restrict__ W0T,         // out: 64*32 floats, W0T[w*32+u]
    float* __restrict__ W1T) {       // out: 32*64 floats, W1T[w*64+u]
  __shared__ float T0[32 * 64];      // Wtp0 @ Wlin0
  __shared__ float T1[64 * 32];      // Wtp1 @ Wlin1
  const int tid = threadIdx.x;

  for (int idx = tid; idx < 32 * 64; idx += 256) {
    int u = idx >> 6, j = idx & 63;
    float s = 0.f;
    for (int k = 0; k < 64; ++k) s += Wtp0[u * 64 + k] * Wlin0[k * 64 + j];
    T0[idx] = s;
  }
  for (int idx = tid; idx < 64 * 32; idx += 256) {
    int u = idx >> 5, j = idx & 31;
    float s = 0.f;
    for (int k = 0; k < 32; ++k) s += Wtp1[u * 32 + k] * Wlin1[k * 32 + j];
    T1[idx] = s;
  }
  __syncthreads();

  const float C0 = 1.0f / (64.0f * sqrtf(96.0f));
  for (int idx = tid; idx < 32 * 64; idx += 256) {
    int u = idx >> 6, w = idx & 63;
    float s = 0.f;
    for (int j = 0; j < 64; ++j) s += T0[u * 64 + j] * Wsc0[j * 64 + w];
    W0T[w * 32 + u] = C0 * s;        // transposed for 8B ds_load B fragments
  }
  const float C1 = 1.0f / 256.0f;
  for (int idx = tid; idx < 64 * 32; idx += 256) {
    int u = idx >> 5, w = idx & 31;
    float s = 0.f;
    for (int j = 0; j < 32; ++j) s += T1[u * 32 + j] * Wsc1[j * 32 + w];
    W1T[w * 64 + u] = C1 * s;
  }
}

// shifted softplus gate: (softplus(x) - log2)/x, x > 0
__device__ __forceinline__ float gate_of(float x) {
  // softplus(x) = x + log1p(exp(-x)) for x > 0 (stable)
  return (x + log1pf(__expf(-x)) - LOG2C) / x;
}

// ---------------------------------------------------------------------------
// Kernel 2: streaming pass. One wave per 16-node tile; two fp32 WMMA GEMMs:
//   P1 (16x32) = X0 (16x64) @ W1           z1/r1 path
//   Z0 (16x64) = S1 (16x32) @ W0           z0/r0 path
// f32 16x16x4 fragment layout (ISA 7.12.2): A lanes 0-15 hold K=k0,k0+1 of
// row M=lane; lanes 16-31 hold K=k0+2,k0+3 of row M=lane-16. B mirrors with
// N=lane%16 striped across lanes. C/D: VGPR j = row j (lanes<16) / j+8.
// ---------------------------------------------------------------------------
__global__ __launch_bounds__(256) void field_layer_kernel(
    const float* __restrict__ node_input, const float* __restrict__ node_attr,
    const float* __restrict__ field, const float* __restrict__ W0T,
    const float* __restrict__ W1T, float* __restrict__ out, int n) {
  __shared__ float sW0T[64 * 32];
  __shared__ float sW1T[32 * 64];
  __shared__ float sFA[TILES_PER_BLOCK][16][4];  // field.xyz, attr per row

  const int tid = threadIdx.x;
  const int wave = tid >> 5;
  const int lane = tid & 31;
  const int r16 = lane & 15;
  const int h = lane >> 4;         // half-wave select (K-pair / row+8)

  for (int i = tid; i < 64 * 32; i += 256) sW0T[i] = W0T[i];
  for (int i = tid; i < 32 * 64; i += 256) sW1T[i] = W1T[i];

  const int tile = blockIdx.x * TILES_PER_BLOCK + wave;
  const int rowBase = tile * 16;
  const bool active = (rowBase + 16) <= n;   // N % 16 == 0 for this problem

  if (active && h == 0) {
    int row = rowBase + r16;
    sFA[wave][r16][0] = field[row * 3 + 0];
    sFA[wave][r16][1] = field[row * 3 + 1];
    sFA[wave][r16][2] = field[row * 3 + 2];
    sFA[wave][r16][3] = node_attr[row];
  }
  __syncthreads();
  if (!active) return;             // EXEC back to all-1s for live waves

  const int row = rowBase + r16;
  const float* __restrict__ xrow = node_input + (size_t)row * ROWLEN;
  const float fldx = sFA[wave][r16][0];
  const float fldy = sFA[wave][r16][1];
  const float fldz = sFA[wave][r16][2];

  // ---- GEMM1: P1 = X0 @ W1  (K=64 -> 16 k-steps, 2 N-tiles) ----
  v8f acc1[2] = {};
#pragma unroll
  for (int kk = 0; kk < 16; ++kk) {
    const int k0 = 4 * kk + 2 * h;                       // even -> 8B aligned
    const v2f a = *(const v2f*)(xrow + k0);
#pragma unroll
    for (int nt = 0; nt < 2; ++nt) {
      const v2f b = *(const v2f*)&sW1T[(16 * nt + r16) * 64 + k0];
      acc1[nt] = __builtin_amdgcn_wmma_f32_16x16x4_f32(
          false, a, false, b, (short)0, acc1[nt], false, false);
    }
  }

  // ---- GEMM0: Z0 = S1 @ W0  (K=32 -> 8 k-steps, 4 N-tiles) ----
  v8f acc0[4] = {};
  const float* __restrict__ x1row = xrow + MUL0;
#pragma unroll
  for (int kk = 0; kk < 8; ++kk) {
    const int u0 = 4 * kk + 2 * h;
    v2f a;
    a.x = x1row[3 * u0 + 0] * fldx + x1row[3 * u0 + 1] * fldy +
          x1row[3 * u0 + 2] * fldz;
    a.y = x1row[3 * u0 + 3] * fldx + x1row[3 * u0 + 4] * fldy +
          x1row[3 * u0 + 5] * fldz;
#pragma unroll
    for (int nt = 0; nt < 4; ++nt) {
      const v2f b = *(const v2f*)&sW0T[(16 * nt + r16) * 32 + u0];
      acc0[nt] = __builtin_amdgcn_wmma_f32_16x16x4_f32(
          false, a, false, b, (short)0, acc0[nt], false, false);
    }
  }

  // ---- Epilogue o0: r0 = x0 + a*Z0; gate; store (64B-coalesced rows) ----
#pragma unroll
  for (int nt = 0; nt < 4; ++nt) {
    const int col = 16 * nt + r16;
#pragma unroll
    for (int j = 0; j < 8; ++j) {
      const int rj = j + 8 * h;
      const size_t grow = (size_t)(rowBase + rj);
      const float av = sFA[wave][rj][3];
      const float r = node_input[grow * ROWLEN + col] + acc0[nt][j] * av;
      const float nn = sqrtf(r * r + EPS2);
      out[grow * ROWLEN + col] = r * gate_of(nn);
    }
  }

  // ---- Epilogue o1: r1 = x1 + a*field⊗P1; vector-norm gate; store ----
#pragma unroll
  for (int nt = 0; nt < 2; ++nt) {
    const int col = 16 * nt + r16;
#pragma unroll
    for (int j = 0; j < 8; ++j) {
      const int rj = j + 8 * h;
      const size_t grow = (size_t)(rowBase + rj);
      const float av = sFA[wave][rj][3];
      const float p = acc1[nt][j] * av;
      const float fx = sFA[wave][rj][0];
      const float fy = sFA[wave][rj][1];
      const float fz = sFA[wave][rj][2];
      const float* __restrict__ xb =
          node_input + grow * ROWLEN + MUL0 + 3 * col;
      const float r0v = xb[0] + fx * p;
      const float r1v = xb[1] + fy * p;
      const float r2v = xb[2] + fz * p;
      const float nn = sqrtf(r0v * r0v + r1v * r1v + r2v * r2v + EPS2);
      const float g = gate_of(nn);
      float* __restrict__ ob = out + grow * ROWLEN + MUL0 + 3 * col;
      ob[0] = r0v * g;
      ob[1] = r1v * g;
      ob[2] = r2v * g;
    }
  }
}

// ---------------------------------------------------------------------------
extern "C" void kernel_launch(void* const* d_in, const int* in_sizes, int n_in,
                              void* d_out, int out_size, void* d_ws,
                              size_t ws_size, hipStream_t stream) {
  const float* node_input = (const float*)d_in[0];
  const float* node_attr  = (const float*)d_in[1];
  const float* field      = (const float*)d_in[2];
  const float* W_tp_0e    = (const float*)d_in[3];
  const float* W_tp_1o    = (const float*)d_in[4];
  const float* W_lin_0e   = (const float*)d_in[5];
  const float* W_lin_1o   = (const float*)d_in[6];
  const float* W_sc_0e    = (const float*)d_in[7];
  const float* W_sc_1o    = (const float*)d_in[8];
  float* out = (float*)d_out;

  float* W0T = (float*)d_ws;     // 64*32 floats
  float* W1T = W0T + 64 * 32;    // 32*64 floats

  const int n = in_sizes[0] / ROWLEN;   // 200000

  fuse_weights_kernel<<<1, 256, 0, stream>>>(W_tp_0e, W_tp_1o, W_lin_0e,
                                             W_lin_1o, W_sc_0e, W_sc_1o, W0T,
                                             W1T);

  const int tiles = (n + 15) / 16;
  const int blocks = (tiles + TILES_PER_BLOCK - 1) / TILES_PER_BLOCK;
  field_layer_kernel<<<blocks, 256, 0, stream>>>(node_input, node_attr, field,
                                                 W0T, W1T, out, n);
}